// PixproLoss_52896817217809
// MI455X (gfx1250) — compile-verified
//
#include <hip/hip_runtime.h>
#include <hip/hip_bf16.h>

// PixPro loss for MI455X (gfx1250, wave32).
// Heavy phase is bandwidth-bound (~205MB @ 23.3TB/s ~= 9us); uses
// V_WMMA_F32_16X16X4_F32 (full fp32) with the channel dim as K and 16
// spatial positions as M/N; diagonals of the 16x16 accumulators give the
// per-position dot / |b|^2 / |m|^2 reductions.

#define NPTS 49      // S*S, S=7
#define CCH  512
#define NBAT 1024

typedef __attribute__((ext_vector_type(2))) float v2f;
typedef __attribute__((ext_vector_type(8))) float v8f;

__device__ __forceinline__ float diag_pick(const v8f a, int r) {
    float v = a[0];
    v = (r == 1) ? a[1] : v;
    v = (r == 2) ? a[2] : v;
    v = (r == 3) ? a[3] : v;
    v = (r == 4) ? a[4] : v;
    v = (r == 5) ? a[5] : v;
    v = (r == 6) ? a[6] : v;
    v = (r == 7) ? a[7] : v;
    return v;
}

// ---------------------------------------------------------------------------
// Kernel 1: cos[b,j] via fp32 WMMA. One block per batch, 4 waves = 4 j-tiles.
// fp32 fragment mapping (16x16x4): lane L, vgpr v -> (M or N) = L%16,
// K = 2*(L/16) + v. Identical for A (MxK) and B (KxN), so a K x 16 tile
// loaded once serves as both operands.
// ---------------------------------------------------------------------------
__global__ __launch_bounds__(128) void cos_wmma_kernel(
    const float* __restrict__ base, const float* __restrict__ moment,
    float* __restrict__ cosw) {
    const int b    = blockIdx.x;
    const int wave = threadIdx.x >> 5;
    const int lane = threadIdx.x & 31;
    const int j0   = wave * 16;
    const int m    = lane & 15;
    const int h    = lane >> 4;          // 0: K={0,1}, 1: K={2,3}
    int j = j0 + m;
    if (j > NPTS - 1) j = NPTS - 1;      // clamp padded lanes (stay in-bounds)

    const float* __restrict__ bp = base   + (size_t)b * (CCH * NPTS);
    const float* __restrict__ mp = moment + (size_t)b * (CCH * NPTS);

    v8f dacc  = {};   // diag -> sum_c b*m
    v8f nbacc = {};   // diag -> sum_c b*b
    v8f nmacc = {};   // diag -> sum_c m*m

#pragma unroll 2
    for (int c0 = 0; c0 < CCH; c0 += 4) {
        const int    c   = c0 + 2 * h;
        const size_t off = (size_t)c * NPTS + j;
        v2f fb, fm;
        fb[0] = bp[off];
        fb[1] = bp[off + NPTS];
        fm[0] = mp[off];
        fm[1] = mp[off + NPTS];
        // prefetch the stride-49 streams a few K-steps ahead (speculative,
        // dropped silently if it runs past the buffer)
        __builtin_prefetch((const void*)(bp + off + 16 * NPTS), 0, 0);
        __builtin_prefetch((const void*)(mp + off + 16 * NPTS), 0, 0);

        dacc  = __builtin_amdgcn_wmma_f32_16x16x4_f32(false, fb, false, fm,
                                                      (short)0, dacc, false, false);
        nbacc = __builtin_amdgcn_wmma_f32_16x16x4_f32(false, fb, false, fb,
                                                      (short)0, nbacc, false, false);
        nmacc = __builtin_amdgcn_wmma_f32_16x16x4_f32(false, fm, false, fm,
                                                      (short)0, nmacc, false, false);
    }

    // Diagonal of 16x16 fp32 C/D: element (d,d) lives in vgpr d%8,
    // lane d (d<8) or lane d+16 (d>=8).
    const int  r     = lane & 7;
    const float dv   = diag_pick(dacc, r);
    const float nb2  = diag_pick(nbacc, r);
    const float nm2  = diag_pick(nmacc, r);
    const bool  lo   = (lane < 8);
    const bool  hi   = (lane >= 24);
    const int   dmi  = lo ? lane : (lane - 16);
    const int   jd   = j0 + dmi;
    if ((lo || hi) && jd < NPTS) {
        const float nb = fmaxf(sqrtf(nb2), 1e-6f);
        const float nm = fmaxf(sqrtf(nm2), 1e-6f);
        cosw[(size_t)b * NPTS + jd] = dv / (nb * nm);
    }
}

// ---------------------------------------------------------------------------
// Kernel 2: per-batch geometry, masks, losses. One block per batch.
// ---------------------------------------------------------------------------
__global__ __launch_bounds__(64) void geom_kernel(
    const float* __restrict__ pbase, const float* __restrict__ pmom,
    const unsigned char* __restrict__ fbase, const unsigned char* __restrict__ fmom,
    const float* __restrict__ cosw,
    float* __restrict__ contrib, float* __restrict__ interf) {
    const int b = blockIdx.x;
    const int t = threadIdx.x;

    __shared__ float cs[NPTS];
    __shared__ float gbx[NPTS], gby[NPTS], gmx[NPTS], gmy[NPTS];
    __shared__ float rsb[64], rnb[64], rsm[64], rnm[64];

    const float x1 = pbase[b * 4 + 0], y1 = pbase[b * 4 + 1];
    const float w1 = pbase[b * 4 + 2], h1 = pbase[b * 4 + 3];
    const float x2 = pmom[b * 4 + 0],  y2 = pmom[b * 4 + 1];
    const float w2 = pmom[b * 4 + 2],  h2 = pmom[b * 4 + 3];
    const bool flb = fbase[b] != 0;
    const bool flm = fmom[b] != 0;

    if (t < NPTS) {
        const int   r  = t / 7;
        const int   q  = t % 7;
        const float tr = (float)r * (1.0f / 6.0f);
        const float tqb = (float)(flb ? (6 - q) : q) * (1.0f / 6.0f);
        const float tqm = (float)(flm ? (6 - q) : q) * (1.0f / 6.0f);
        gbx[t] = x1 + w1 * tr;
        gby[t] = y1 + h1 * tqb;
        gmx[t] = x2 + w2 * tr;
        gmy[t] = y2 + h2 * tqm;
        cs[t]  = cosw[(size_t)b * NPTS + t];
    }
    __syncthreads();

    const float tb = 0.7f * sqrtf(w1 * w1 + h1 * h1);
    const float tm = 0.7f * sqrtf(w2 * w2 + h2 * h2);

    float sb = 0.f, nnzb = 0.f, sm = 0.f, nnzm = 0.f;
    if (t < NPTS) {
        const float gx = gbx[t], gy = gby[t], ci = cs[t];
#pragma unroll 7
        for (int j = 0; j < NPTS; ++j) {
            const float dx = gx - gmx[j];
            const float dy = gy - gmy[j];
            const float d  = sqrtf(dx * dx + dy * dy);
            if (d < tb) { sb += cs[j]; nnzb += 1.0f; }
            if (d < tm) { sm += ci;    nnzm += 1.0f; }
        }
    }
    rsb[t] = sb; rnb[t] = nnzb; rsm[t] = sm; rnm[t] = nnzm;
    __syncthreads();
    for (int s = 32; s > 0; s >>= 1) {
        if (t < s) {
            rsb[t] += rsb[t + s];
            rnb[t] += rnb[t + s];
            rsm[t] += rsm[t + s];
            rnm[t] += rnm[t + s];
        }
        __syncthreads();
    }
    if (t == 0) {
        const float lossb = (rnb[0] > 0.f) ? (rsb[0] / fmaxf(rnb[0], 1.0f)) : 0.f;
        const float lossm = (rnm[0] > 0.f) ? (rsm[0] / fmaxf(rnm[0], 1.0f)) : 0.f;
        const float cx1 = x1 + 0.5f * w1, cx2 = x2 + 0.5f * w2;
        const float cy1 = y1 + 0.5f * h1, cy2 = y2 + 0.5f * h2;
        const bool inter = (fabsf(cx1 - cx2) * 2.f < (w1 + w2)) &&
                           (fabsf(cy1 - cy2) * 2.f < (h1 + h2));
        contrib[b] = inter ? -(lossb + lossm) : 0.f;
        interf[b]  = inter ? 1.f : 0.f;
    }
}

// ---------------------------------------------------------------------------
// Kernel 3: final scalar reduce (no atomics -> no ws zero-init needed).
// ---------------------------------------------------------------------------
__global__ __launch_bounds__(1024) void final_reduce_kernel(
    const float* __restrict__ contrib, const float* __restrict__ interf,
    float* __restrict__ out) {
    __shared__ float ss[1024];
    __shared__ float sc[1024];
    const int t = threadIdx.x;
    ss[t] = contrib[t];
    sc[t] = interf[t];
    __syncthreads();
    for (int s = 512; s > 0; s >>= 1) {
        if (t < s) { ss[t] += ss[t + s]; sc[t] += sc[t + s]; }
        __syncthreads();
    }
    if (t == 0) out[0] = ss[0] / fmaxf(sc[0], 1.0f);
}

extern "C" void kernel_launch(void* const* d_in, const int* in_sizes, int n_in,
                              void* d_out, int out_size, void* d_ws, size_t ws_size,
                              hipStream_t stream) {
    const float* base   = (const float*)d_in[0];
    const float* moment = (const float*)d_in[1];
    const float* pbase  = (const float*)d_in[2];
    const float* pmom   = (const float*)d_in[3];
    // JAX bool arrays are 1 byte/elem; flag b is read as byte b.
    const unsigned char* fbase = (const unsigned char*)d_in[4];
    const unsigned char* fmom  = (const unsigned char*)d_in[5];

    float* ws      = (float*)d_ws;
    float* cosw    = ws;                       // 1024*49 floats
    float* contrib = ws + (size_t)NBAT * NPTS; // 1024 floats
    float* interf  = contrib + NBAT;           // 1024 floats

    cos_wmma_kernel<<<NBAT, 128, 0, stream>>>(base, moment, cosw);
    geom_kernel<<<NBAT, 64, 0, stream>>>(pbase, pmom, fbase, fmom, cosw,
                                         contrib, interf);
    final_reduce_kernel<<<1, 1024, 0, stream>>>(contrib, interf, (float*)d_out);
}